// ConfigurationLXMERTDecoder_55018531061886
// MI455X (gfx1250) — compile-verified
//
#include <hip/hip_runtime.h>
#include <hip/hip_bf16.h>
#include <math.h>

// ---------------- problem constants ----------------
#define B_    64
#define S_    80
#define L_    8
#define C_    12
#define OBJ_  36
#define AFS_  4
#define E_    64
#define H_    512
#define FEAT_ 2052
#define TOPN_ 3
#define OBJD_ 2048
#define XDIM_ (E_ + FEAT_)            // 2116
#define G4H_  (4 * H_)                // 2048
#define HC_   (2 * H_)                // 1024
#define CANDD_ (FEAT_ + TOPN_ * OBJD_) // 8196
#define SL_   (S_ * L_)               // 640

typedef __attribute__((ext_vector_type(2))) float v2f;
typedef __attribute__((ext_vector_type(8))) float v8f;

// =====================================================================
// Generic WMMA fp32 GEMM:  Out[m,n] (+)= sum_k A[m,k] * W[n,k]
// M fixed at 64 (4 tiles of 16). One wave computes a 16-wide N strip for
// all 64 rows with V_WMMA_F32_16X16X4_F32, reusing the B fragment across
// the 4 M tiles. Software-pipelined: fragments for k+4 are issued before
// the WMMAs consuming the k fragments, so each load gets a full
// iteration of latency slack (grid parallelism alone is too small on
// MI455X to hide HBM latency for these modest-N GEMMs).
// K must be a multiple of 4 and lda/ldw even (true at all call sites),
// so float2 fragment loads are 8B aligned.
// =====================================================================
__global__ void __launch_bounds__(128)
gemm_xwt_wmma(const float* __restrict__ A, int lda,
              const float* __restrict__ W, int ldw,
              float* __restrict__ Out, int ldo,
              const float* __restrict__ bias0,
              const float* __restrict__ bias1,
              int N, int K, int accumulate, int do_tanh)
{
    const int wave = threadIdx.x >> 5;
    const int n0   = (blockIdx.x * 4 + wave) * 16;
    if (n0 >= N) return;                       // uniform per wave

    const int lane = threadIdx.x & 31;
    const int hf   = lane >> 4;                // 0: K=0,1   1: K=2,3
    const int lr   = lane & 15;

    const int  nB  = n0 + lr;                  // W row this lane feeds into B-fragment
    const bool nOK = (nB < N);
    const float* Wrow = W + (size_t)(nOK ? nB : 0) * (size_t)ldw;

    const float* Ar0 = A + (size_t)(lr +  0) * (size_t)lda;
    const float* Ar1 = A + (size_t)(lr + 16) * (size_t)lda;
    const float* Ar2 = A + (size_t)(lr + 32) * (size_t)lda;
    const float* Ar3 = A + (size_t)(lr + 48) * (size_t)lda;

    v8f acc0 = {0.f,0.f,0.f,0.f,0.f,0.f,0.f,0.f};
    v8f acc1 = acc0, acc2 = acc0, acc3 = acc0;
    const v2f zz = {0.f, 0.f};

    // ---- pipelined K loop: preload k=0, prefetch k+4 ahead of WMMAs ----
    int kk = hf * 2;
    v2f bv = *(const v2f*)(Wrow + kk); if (!nOK) bv = zz;
    v2f a0 = *(const v2f*)(Ar0 + kk);
    v2f a1 = *(const v2f*)(Ar1 + kk);
    v2f a2 = *(const v2f*)(Ar2 + kk);
    v2f a3 = *(const v2f*)(Ar3 + kk);

    for (int k = 4; k < K; k += 4) {
        const int kn = k + hf * 2;
        v2f bvn = *(const v2f*)(Wrow + kn); if (!nOK) bvn = zz;
        v2f a0n = *(const v2f*)(Ar0 + kn);
        v2f a1n = *(const v2f*)(Ar1 + kn);
        v2f a2n = *(const v2f*)(Ar2 + kn);
        v2f a3n = *(const v2f*)(Ar3 + kn);
        acc0 = __builtin_amdgcn_wmma_f32_16x16x4_f32(false, a0, false, bv, (short)0, acc0, false, false);
        acc1 = __builtin_amdgcn_wmma_f32_16x16x4_f32(false, a1, false, bv, (short)0, acc1, false, false);
        acc2 = __builtin_amdgcn_wmma_f32_16x16x4_f32(false, a2, false, bv, (short)0, acc2, false, false);
        acc3 = __builtin_amdgcn_wmma_f32_16x16x4_f32(false, a3, false, bv, (short)0, acc3, false, false);
        bv = bvn; a0 = a0n; a1 = a1n; a2 = a2n; a3 = a3n;
    }
    // drain last stage
    acc0 = __builtin_amdgcn_wmma_f32_16x16x4_f32(false, a0, false, bv, (short)0, acc0, false, false);
    acc1 = __builtin_amdgcn_wmma_f32_16x16x4_f32(false, a1, false, bv, (short)0, acc1, false, false);
    acc2 = __builtin_amdgcn_wmma_f32_16x16x4_f32(false, a2, false, bv, (short)0, acc2, false, false);
    acc3 = __builtin_amdgcn_wmma_f32_16x16x4_f32(false, a3, false, bv, (short)0, acc3, false, false);

    const int nOut = n0 + lr;                  // C/D: lane -> column
    if (nOut >= N) return;
    float bsum = 0.f;
    if (bias0) bsum += bias0[nOut];
    if (bias1) bsum += bias1[nOut];

    #pragma unroll
    for (int r = 0; r < 8; ++r) {
        const int mloc = hf * 8 + r;           // C/D: vgpr r, half -> row within tile
        float v0 = acc0[r], v1 = acc1[r], v2 = acc2[r], v3 = acc3[r];
        float* p0 = Out + (size_t)(mloc +  0) * (size_t)ldo + nOut;
        float* p1 = Out + (size_t)(mloc + 16) * (size_t)ldo + nOut;
        float* p2 = Out + (size_t)(mloc + 32) * (size_t)ldo + nOut;
        float* p3 = Out + (size_t)(mloc + 48) * (size_t)ldo + nOut;
        v0 += bsum; v1 += bsum; v2 += bsum; v3 += bsum;
        if (accumulate) { v0 += *p0; v1 += *p1; v2 += *p2; v3 += *p3; }
        if (do_tanh) { v0 = tanhf(v0); v1 = tanhf(v1); v2 = tanhf(v2); v3 = tanhf(v3); }
        *p0 = v0; *p1 = v1; *p2 = v2; *p3 = v3;
    }
}

// ---------------- block reduce helper ----------------
__device__ __forceinline__ float block_reduce_sum256(float v, float* sbuf) {
    const int t = threadIdx.x;
    sbuf[t] = v; __syncthreads();
    for (int s = 128; s > 0; s >>= 1) {
        if (t < s) sbuf[t] += sbuf[t + s];
        __syncthreads();
    }
    float r = sbuf[0]; __syncthreads();
    return r;
}

// =====================================================================
// Feature attention: softmax over OBJ_ dots(2052), weighted sum written
// into the x-concat buffer at column offset E_.
// =====================================================================
__global__ void __launch_bounds__(256)
feat_att_kernel(const float* __restrict__ feature,   // [B,36,2052]
                const float* __restrict__ tgt_f,     // [B,2052]
                float* __restrict__ xcat)            // [B,2116]
{
    __shared__ float sbuf[256];
    __shared__ float sattn[OBJ_];
    const int b = blockIdx.x, tid = threadIdx.x;
    const float* fb = feature + (size_t)b * OBJ_ * FEAT_;
    const float* tb = tgt_f + (size_t)b * FEAT_;

    for (int s = 0; s < OBJ_; ++s) {
        float p = 0.f;
        for (int d = tid; d < FEAT_; d += 256) p += fb[(size_t)s * FEAT_ + d] * tb[d];
        float tot = block_reduce_sum256(p, sbuf);
        if (tid == 0) sattn[s] = tot;
    }
    __syncthreads();
    if (tid == 0) {
        float mx = sattn[0];
        for (int s = 1; s < OBJ_; ++s) mx = fmaxf(mx, sattn[s]);
        float sum = 0.f;
        for (int s = 0; s < OBJ_; ++s) { sattn[s] = expf(sattn[s] - mx); sum += sattn[s]; }
        float inv = 1.f / sum;
        for (int s = 0; s < OBJ_; ++s) sattn[s] *= inv;
    }
    __syncthreads();
    float* ob = xcat + (size_t)b * XDIM_ + E_;
    for (int d = tid; d < FEAT_; d += 256) {
        float acc = 0.f;
        for (int s = 0; s < OBJ_; ++s) acc += sattn[s] * fb[(size_t)s * FEAT_ + d];
        ob[d] = acc;
    }
}

// =====================================================================
// LSTM cell elementwise: gates order [i,f,g,o]. Writes h1,c1 to d_out
// and h1 also into h-concat columns [512,1024).
// =====================================================================
__global__ void __launch_bounds__(256)
lstm_kernel(const float* __restrict__ gates, const float* __restrict__ c0,
            float* __restrict__ h1_out, float* __restrict__ c1_out,
            float* __restrict__ hcat)
{
    const int i = blockIdx.x * 256 + threadIdx.x;
    if (i >= B_ * H_) return;
    const int b = i >> 9, h = i & (H_ - 1);
    const float* g = gates + (size_t)b * G4H_;
    const float gi = g[h], gf = g[H_ + h], gg = g[2 * H_ + h], go = g[3 * H_ + h];
    const float si = 1.f / (1.f + expf(-gi));
    const float sf = 1.f / (1.f + expf(-gf));
    const float so = 1.f / (1.f + expf(-go));
    const float c1 = sf * c0[i] + si * tanhf(gg);
    const float h1 = so * tanhf(c1);
    c1_out[i] = c1;
    h1_out[i] = h1;
    hcat[(size_t)b * HC_ + H_ + h] = h1;
}

// =====================================================================
// Context attention: masked softmax over S_, emits ctx_attn (output) and
// wctx into h-concat columns [0,512).
// =====================================================================
__global__ void __launch_bounds__(256)
ctx_att_kernel(const float* __restrict__ ctx,        // [B,80,512]
               const float* __restrict__ tgt_c,      // [B,512]
               const unsigned char* __restrict__ ctx_mask, // [B,80] bool
               float* __restrict__ ctx_attn_out,     // [B,80]
               float* __restrict__ hcat)             // [B,1024]
{
    __shared__ float sbuf[256];
    __shared__ float sattn[S_];
    const int b = blockIdx.x, tid = threadIdx.x;
    const float* tb = tgt_c + (size_t)b * H_;

    for (int s = 0; s < S_; ++s) {
        const float* cb = ctx + ((size_t)b * S_ + s) * H_;
        float p = 0.f;
        for (int h = tid; h < H_; h += 256) p += cb[h] * tb[h];
        float tot = block_reduce_sum256(p, sbuf);
        if (tid == 0) sattn[s] = ctx_mask[(size_t)b * S_ + s] ? -INFINITY : tot;
    }
    __syncthreads();
    if (tid == 0) {
        float mx = -INFINITY;
        for (int s = 0; s < S_; ++s) mx = fmaxf(mx, sattn[s]);
        float sum = 0.f;
        for (int s = 0; s < S_; ++s) { float e = expf(sattn[s] - mx); sattn[s] = e; sum += e; }
        float inv = 1.f / sum;
        for (int s = 0; s < S_; ++s) sattn[s] *= inv;
    }
    __syncthreads();
    for (int s = tid; s < S_; s += 256) ctx_attn_out[(size_t)b * S_ + s] = sattn[s];
    for (int h = tid; h < H_; h += 256) {
        float acc = 0.f;
        for (int s = 0; s < S_; ++s) acc += sattn[s] * ctx[((size_t)b * S_ + s) * H_ + h];
        hcat[(size_t)b * HC_ + h] = acc;
    }
}

// =====================================================================
// Stable top-3 of scores[j] = ctx_attn[b, j/8] * landmark_mask[b, j]
// (ties -> smaller index first, matching stable argsort of -scores).
// =====================================================================
__global__ void __launch_bounds__(256)
top3_kernel(const float* __restrict__ ctx_attn,       // [B,80]
            const float* __restrict__ landmark_mask,  // [B,640]
            int* __restrict__ top_idx)                // [B,3]
{
    __shared__ float sv[256];
    __shared__ int   si[256];
    __shared__ int   ssel[TOPN_];
    const int b = blockIdx.x, tid = threadIdx.x;

    for (int r = 0; r < TOPN_; ++r) {
        float bv = -INFINITY; int bi = SL_;
        for (int j = tid; j < SL_; j += 256) {
            bool excl = false;
            for (int q = 0; q < r; ++q) if (ssel[q] == j) excl = true;
            if (excl) continue;
            const float v = ctx_attn[(size_t)b * S_ + (j >> 3)] * landmark_mask[(size_t)b * SL_ + j];
            if (v > bv || (v == bv && j < bi)) { bv = v; bi = j; }
        }
        sv[tid] = bv; si[tid] = bi; __syncthreads();
        for (int st = 128; st > 0; st >>= 1) {
            if (tid < st) {
                if (sv[tid + st] > sv[tid] || (sv[tid + st] == sv[tid] && si[tid + st] < si[tid])) {
                    sv[tid] = sv[tid + st]; si[tid] = si[tid + st];
                }
            }
            __syncthreads();
        }
        if (tid == 0) { ssel[r] = si[0]; top_idx[b * TOPN_ + r] = si[0]; }
        __syncthreads();
    }
}

// =====================================================================
// Final logits: dot(cand_full[b,c,:], tgt_k[b,:]) without materializing
// cand_full — gathers the 3 selected object rows on the fly.
// =====================================================================
__global__ void __launch_bounds__(256)
logit_kernel(const float* __restrict__ cand_feat,    // [B,12,2052]
             const float* __restrict__ cof,          // [B,12,36,2048]
             const long long* __restrict__ sim,      // [2,B,12,640]
             const int* __restrict__ top_idx,        // [B,3]
             const float* __restrict__ tgt_k,        // [B,8196]
             float* __restrict__ logit_out)          // [B,12]
{
    __shared__ float sbuf[256];
    __shared__ int   sidx[TOPN_];
    const int bc = blockIdx.x;
    const int b = bc / C_, c = bc % C_;
    const int tid = threadIdx.x;

    if (tid < TOPN_) {
        const long long v = sim[((size_t)(B_ * C_) + (size_t)b * C_ + c) * SL_ + top_idx[b * TOPN_ + tid]];
        sidx[tid] = (int)v;
    }
    __syncthreads();

    const float* tk = tgt_k + (size_t)b * CANDD_;
    const float* cf = cand_feat + ((size_t)b * C_ + c) * FEAT_;
    float p = 0.f;
    for (int d = tid; d < FEAT_; d += 256) p += cf[d] * tk[d];
    for (int n = 0; n < TOPN_; ++n) {
        const float* of  = cof + (((size_t)b * C_ + c) * OBJ_ + sidx[n]) * OBJD_;
        const float* tk2 = tk + FEAT_ + n * OBJD_;
        for (int d = tid; d < OBJD_; d += 256) p += of[d] * tk2[d];
    }
    const float tot = block_reduce_sum256(p, sbuf);
    if (tid == 0) logit_out[bc] = tot;
}

// =====================================================================
// Host-side orchestration
// =====================================================================
extern "C" void kernel_launch(void* const* d_in, const int* in_sizes, int n_in,
                              void* d_out, int out_size, void* d_ws, size_t ws_size,
                              hipStream_t stream) {
    (void)in_sizes; (void)n_in; (void)out_size; (void)ws_size;

    const float* action    = (const float*)d_in[0];
    const float* feature   = (const float*)d_in[1];
    const float* cand_feat = (const float*)d_in[2];
    /* d_in[3] = h_0 (unused by reference) */
    const float* prev_h1   = (const float*)d_in[4];
    const float* c_0       = (const float*)d_in[5];
    const float* ctx       = (const float*)d_in[6];
    /* d_in[7] = step, d_in[8] = s_0, d_in[9] = r_t : unused */
    const unsigned char* ctx_mask = (const unsigned char*)d_in[10];
    const float* landmark_mask    = (const float*)d_in[11];
    /* d_in[12] = candi_landmark : unused */
    const float* cof       = (const float*)d_in[13];
    const long long* sim   = (const long long*)d_in[14];
    const float* W_emb     = (const float*)d_in[15];
    const float* b_emb     = (const float*)d_in[16];
    const float* W_ih      = (const float*)d_in[17];
    const float* b_ih      = (const float*)d_in[18];
    const float* W_hh      = (const float*)d_in[19];
    const float* b_hh      = (const float*)d_in[20];
    const float* W_in_feat = (const float*)d_in[21];
    const float* W_in_ctx  = (const float*)d_in[22];
    const float* W_out_ctx = (const float*)d_in[23];
    const float* W_in_cand = (const float*)d_in[24];

    // output layout (floats): h_1 | c_1 | logit | h_tilde | ctx_attn
    float* out       = (float*)d_out;
    float* o_h1      = out;
    float* o_c1      = out + (size_t)B_ * H_;
    float* o_logit   = out + (size_t)2 * B_ * H_;
    float* o_htilde  = out + (size_t)2 * B_ * H_ + (size_t)B_ * C_;
    float* o_ctxattn = o_htilde + (size_t)B_ * H_;

    // workspace layout (floats)
    float* ws      = (float*)d_ws;
    float* w_tgtf  = ws;                                   // 64*2052
    float* w_xcat  = w_tgtf + (size_t)B_ * FEAT_;          // 64*2116
    float* w_gates = w_xcat + (size_t)B_ * XDIM_;          // 64*2048
    float* w_tgtc  = w_gates + (size_t)B_ * G4H_;          // 64*512
    float* w_hcat  = w_tgtc + (size_t)B_ * H_;             // 64*1024
    float* w_tgtk  = w_hcat + (size_t)B_ * HC_;            // 64*8196
    int*   w_top   = (int*)(w_tgtk + (size_t)B_ * CANDD_); // 64*3 ints

    auto gblocks = [](int N) { return (((N + 15) / 16) + 3) / 4; };

    // 1) action embedding -> xcat[:, 0:64], tanh, +b_emb
    gemm_xwt_wmma<<<gblocks(E_), 128, 0, stream>>>(
        action, AFS_, W_emb, AFS_, w_xcat, XDIM_, b_emb, nullptr, E_, AFS_, 0, 1);

    // 2) tgt_f = prev_h1 @ W_in_feat^T
    gemm_xwt_wmma<<<gblocks(FEAT_), 128, 0, stream>>>(
        prev_h1, H_, W_in_feat, H_, w_tgtf, FEAT_, nullptr, nullptr, FEAT_, H_, 0, 0);

    // 3) feature attention -> xcat[:, 64:2116]
    feat_att_kernel<<<B_, 256, 0, stream>>>(feature, w_tgtf, w_xcat);

    // 4) gates = xcat @ W_ih^T + b_ih + b_hh
    gemm_xwt_wmma<<<gblocks(G4H_), 128, 0, stream>>>(
        w_xcat, XDIM_, W_ih, XDIM_, w_gates, G4H_, b_ih, b_hh, G4H_, XDIM_, 0, 0);

    // 5) gates += prev_h1 @ W_hh^T
    gemm_xwt_wmma<<<gblocks(G4H_), 128, 0, stream>>>(
        prev_h1, H_, W_hh, H_, w_gates, G4H_, nullptr, nullptr, G4H_, H_, 1, 0);

    // 6) LSTM elementwise -> h1, c1 (also h1 into hcat[:,512:1024])
    lstm_kernel<<<(B_ * H_) / 256, 256, 0, stream>>>(w_gates, c_0, o_h1, o_c1, w_hcat);

    // 7) tgt_c = h1 @ W_in_ctx^T
    gemm_xwt_wmma<<<gblocks(H_), 128, 0, stream>>>(
        o_h1, H_, W_in_ctx, H_, w_tgtc, H_, nullptr, nullptr, H_, H_, 0, 0);

    // 8) ctx attention -> ctx_attn out + wctx into hcat[:,0:512]
    ctx_att_kernel<<<B_, 256, 0, stream>>>(ctx, w_tgtc, ctx_mask, o_ctxattn, w_hcat);

    // 9) h_tilde = tanh(hcat @ W_out_ctx^T)
    gemm_xwt_wmma<<<gblocks(H_), 128, 0, stream>>>(
        w_hcat, HC_, W_out_ctx, HC_, o_htilde, H_, nullptr, nullptr, H_, HC_, 0, 1);

    // 10) stable top-3 indices
    top3_kernel<<<B_, 256, 0, stream>>>(o_ctxattn, landmark_mask, w_top);

    // 11) tgt_k = h_tilde @ W_in_cand^T
    gemm_xwt_wmma<<<gblocks(CANDD_), 128, 0, stream>>>(
        o_htilde, H_, W_in_cand, H_, w_tgtk, CANDD_, nullptr, nullptr, CANDD_, H_, 0, 0);

    // 12) gathered candidate logits
    logit_kernel<<<B_ * C_, 256, 0, stream>>>(cand_feat, cof, sim, w_top, w_tgtk, o_logit);
}